// IntraSentenceAttention_27728308863582
// MI455X (gfx1250) — compile-verified
//
#include <hip/hip_runtime.h>
#include <hip/hip_bf16.h>

// IntraSentenceAttention for MI455X (gfx1250, wave32, WMMA).
// Single-pass "flash" formulation is exact: the reference uses raw exp() with
// no max-subtraction, so out = (sum_s e_ts * x_s) / (sum_s e_ts + eps) can be
// accumulated streaming. Both GEMMs use native fp32 WMMA
// (V_WMMA_F32_16X16X4_F32) to match fp32 reference precision; total work is
// ~17 GFLOP and the whole 16.7 MB working set is L2-resident, so the kernel
// is matrix-issue bound -> maximize WMMA burst density and hide tile loads
// with register double-buffering.

typedef float v2f __attribute__((ext_vector_type(2)));
typedef float v8f __attribute__((ext_vector_type(8)));

#define TSEQ 1024
#define DDIM 128
#define KEPS 1e-7f
#define XSTR 132   // 128 + 4 floats pad -> conflict-free LDS banking
#define ESTR 17    // 16 + 1 pad

__launch_bounds__(256)
__global__ void isa_attn_kernel(const float* __restrict__ x,
                                const int*   __restrict__ mask,
                                float*       __restrict__ out)
{
    __shared__ float xc[2][16 * XSTR];          // double-buffered key tile
    __shared__ float cmask[2][16];              // double-buffered key mask
    __shared__ float e_lds[8][16 * ESTR];       // per-wave 16x16 E tile

    const int tid  = threadIdx.x;
    const int lane = tid & 31;
    const int wave = tid >> 5;
    const int b    = blockIdx.x >> 3;           // 32 batches
    const int rblk = blockIdx.x & 7;            // 8 row-blocks of 128 rows
    const int r0   = rblk * 128 + wave * 16;    // this wave's first query row

    const float* xb = x    + (size_t)b * TSEQ * DDIM;
    const int*   mb = mask + (size_t)b * TSEQ;
    float*       ob = out  + (size_t)b * TSEQ * DDIM;

    const int l16   = lane & 15;                // M (A-frag) / N (B,C-frag)
    const int khalf = (lane >> 4) * 2;          // K offset of this lane-half
    const int mhalf = (lane >> 4) * 8;          // M offset of this lane-half

    // Staging map: each thread owns float4 units (tid) and (tid + 256).
    const int srow0 = tid >> 5,          scol0 = tid & 31;
    const int srow1 = (tid + 256) >> 5,  scol1 = (tid + 256) & 31;

    // ---- Preload all A fragments for this wave's 16 query rows (K = 128) ----
    // A-frag (16x4 f32): lane l holds (m = l%16, k = kc*4 + (l/16)*2 + {0,1}).
    v2f afrag[32];
    {
        const float* arow = xb + (size_t)(r0 + l16) * DDIM + khalf;
        #pragma unroll
        for (int kc = 0; kc < 32; ++kc) {
            afrag[kc][0] = arow[kc * 4 + 0];
            afrag[kc][1] = arow[kc * 4 + 1];
        }
    }
    float mrow[8];
    #pragma unroll
    for (int i = 0; i < 8; ++i)
        mrow[i] = mb[r0 + mhalf + i] ? 1.0f : 0.0f;

    v8f zero = {};
    v8f acc[8];                                 // 16x128 output accumulator
    #pragma unroll
    for (int dn = 0; dn < 8; ++dn) acc[dn] = zero;
    float rs[8];                                // raw row-sum accumulators
    #pragma unroll
    for (int i = 0; i < 8; ++i) rs[i] = 0.0f;

    // ---- prologue: stage key tile 0 ----
    {
        float4 a = ((const float4*)(xb + srow0 * DDIM))[scol0];
        float4 c = ((const float4*)(xb + srow1 * DDIM))[scol1];
        float* d0 = &xc[0][srow0 * XSTR + scol0 * 4];
        d0[0] = a.x; d0[1] = a.y; d0[2] = a.z; d0[3] = a.w;
        float* d1 = &xc[0][srow1 * XSTR + scol1 * 4];
        d1[0] = c.x; d1[1] = c.y; d1[2] = c.z; d1[3] = c.w;
        if (tid < 16) cmask[0][tid] = mb[tid] ? 1.0f : 0.0f;
    }
    __syncthreads();

    int cur = 0;
    for (int ct = 0; ct < 64; ++ct) {
        // ---- issue global loads for tile ct+1 early (latency hidden by GEMMs) ----
        float4 n0, n1; float ncm = 0.0f;
        if (ct < 63) {
            const float* src = xb + (size_t)(ct + 1) * 16 * DDIM;
            n0 = ((const float4*)(src + srow0 * DDIM))[scol0];
            n1 = ((const float4*)(src + srow1 * DDIM))[scol1];
            if (tid < 16) ncm = mb[(ct + 1) * 16 + tid] ? 1.0f : 0.0f;
        }

        const float* xt = &xc[cur][0];

        // ---- scores: S(16x16) = A(16x128) * Xc^T, grouped 8-WMMA bursts ----
        // B-frag (4x16 f32): lane l holds (k = kc*4 + (l/16)*2 + {0,1}, n = l%16);
        // B[k][n] = Xc[n][k] -> contiguous float2 in the padded LDS row.
        v8f s0 = zero, s1 = zero;               // two chains shorten dep chain
        const float* bcol = xt + l16 * XSTR + khalf;
        #pragma unroll
        for (int g = 0; g < 4; ++g) {
            v2f bf[8];                          // distinct regs -> batched ds loads
            #pragma unroll
            for (int j = 0; j < 8; ++j) {
                int kc = g * 8 + j;
                bf[j][0] = bcol[kc * 4 + 0];
                bf[j][1] = bcol[kc * 4 + 1];
            }
            #pragma unroll
            for (int j = 0; j < 8; j += 2) {
                int kc = g * 8 + j;
                s0 = __builtin_amdgcn_wmma_f32_16x16x4_f32(false, afrag[kc],     false, bf[j],
                                                           (short)0, s0, false, false);
                s1 = __builtin_amdgcn_wmma_f32_16x16x4_f32(false, afrag[kc + 1], false, bf[j + 1],
                                                           (short)0, s1, false, false);
            }
        }
        v8f s = s0 + s1;

        // ---- epilogue: e = exp(s + min(i-j,10)) * mrow * mcol; spill to LDS ----
        const float mcol = cmask[cur][l16];
        const int   gj   = ct * 16 + l16;
        float* ew = &e_lds[wave][0];
        #pragma unroll
        for (int i = 0; i < 8; ++i) {
            int   m    = mhalf + i;             // C-frag: vgpr i -> M = i + 8*(lane>=16)
            float dist = fminf((float)(r0 + m - gj), 10.0f);  // no lower clamp (matches ref)
            float e    = __expf(s[i] + dist) * mrow[i] * mcol;
            rs[i] += e;
            ew[m * ESTR + l16] = e;
        }
        asm volatile("s_wait_dscnt 0" ::: "memory");  // wave-local LDS RAW ordering

        // ---- acc(16x128) += E(16x16) @ Xc(16x128): 4 K-chunks x 8-WMMA bursts ----
        #pragma unroll
        for (int kc2 = 0; kc2 < 4; ++kc2) {
            v2f a2;                             // A[m][k] = E[m][k], contiguous float2
            a2[0] = ew[l16 * ESTR + kc2 * 4 + khalf + 0];
            a2[1] = ew[l16 * ESTR + kc2 * 4 + khalf + 1];
            v2f bb[8];                          // B[k][d] = Xc[k][dn*16 + n]
            #pragma unroll
            for (int dn = 0; dn < 8; ++dn) {
                bb[dn][0] = xt[(kc2 * 4 + khalf + 0) * XSTR + dn * 16 + l16];
                bb[dn][1] = xt[(kc2 * 4 + khalf + 1) * XSTR + dn * 16 + l16];
            }
            #pragma unroll
            for (int dn = 0; dn < 8; ++dn)      // 8 independent accumulation chains
                acc[dn] = __builtin_amdgcn_wmma_f32_16x16x4_f32(false, a2, false, bb[dn],
                                                                (short)0, acc[dn],
                                                                false, false);
        }

        // ---- commit staged tile ct+1 into the other buffer ----
        if (ct < 63) {
            float* d0 = &xc[cur ^ 1][srow0 * XSTR + scol0 * 4];
            d0[0] = n0.x; d0[1] = n0.y; d0[2] = n0.z; d0[3] = n0.w;
            float* d1 = &xc[cur ^ 1][srow1 * XSTR + scol1 * 4];
            d1[0] = n1.x; d1[1] = n1.y; d1[2] = n1.z; d1[3] = n1.w;
            if (tid < 16) cmask[cur ^ 1][tid] = ncm;
        }
        __syncthreads();
        cur ^= 1;
    }

    // ---- reduce row sums across each 16-lane half, normalize, store ----
    #pragma unroll
    for (int i = 0; i < 8; ++i) {
        float v = rs[i];
        v += __shfl_xor(v, 1);                  // xor masks 1..8 stay within the
        v += __shfl_xor(v, 2);                  // 16-lane half owning row mhalf+i
        v += __shfl_xor(v, 4);
        v += __shfl_xor(v, 8);
        rs[i] = 1.0f / (v + KEPS);
    }
    #pragma unroll
    for (int dn = 0; dn < 8; ++dn) {
        #pragma unroll
        for (int i = 0; i < 8; ++i) {
            int m = mhalf + i;
            ob[(size_t)(r0 + m) * DDIM + dn * 16 + l16] = acc[dn][i] * rs[i];
        }
    }
}

extern "C" void kernel_launch(void* const* d_in, const int* in_sizes, int n_in,
                              void* d_out, int out_size, void* d_ws, size_t ws_size,
                              hipStream_t stream) {
    (void)in_sizes; (void)n_in; (void)out_size; (void)d_ws; (void)ws_size;
    const float* x    = (const float*)d_in[0];          // [32,1024,128] fp32
    const int*   mask = (const int*)d_in[1];            // [32,1024] (integer/bool)
    float*       out  = (float*)d_out;                  // [32,1024,128] fp32
    // 32 batches x 8 row-blocks of 128 query rows; 8 wave32 waves per block.
    isa_attn_kernel<<<dim3(256), dim3(256), 0, stream>>>(x, mask, out);
}